// SwinTransformer_TransInMo_57612691309528
// MI455X (gfx1250) — compile-verified
//
#include <hip/hip_runtime.h>

typedef __attribute__((ext_vector_type(2))) float v2f;
typedef __attribute__((ext_vector_type(4))) float v4f;
typedef __attribute__((ext_vector_type(8))) float v8f;

#define WMMA_F32(a, b, c) \
  __builtin_amdgcn_wmma_f32_16x16x4_f32(false, (a), false, (b), (short)0, (c), false, false)

constexpr int B_IMG = 32, HEADS = 6, NWIN = 64, NWIN_T = 16;
constexpr int NTOK = 49, HD = 32, CDIM = 192, DDIM = NTOK * HD; // 1568
constexpr int BWIN = B_IMG * NWIN;                              // 2048
constexpr int TOK = BWIN * NTOK;                                // 100352
constexpr float SCALE = 0.17677669529663687f;                   // 32^-0.5

// ---------------------------------------------------------------------------
// Kernel 0: expand relative-position bias: biasx[h][m][n] = table[ridx[m,n], h]
// ---------------------------------------------------------------------------
__global__ void k_bias_expand(const float* __restrict__ table,
                              const int* __restrict__ ridx,
                              float* __restrict__ biasx) {
  int i = blockIdx.x * blockDim.x + threadIdx.x;
  if (i >= HEADS * NTOK * NTOK) return;
  int h = i / (NTOK * NTOK);
  int mn = i - h * (NTOK * NTOK);
  biasx[i] = table[ridx[mn] * HEADS + h];
}

// ---------------------------------------------------------------------------
// Kernel 1: template cross-attention. One block per (b,h); 4 waves, one
// 16-window m-tile each.  L = (X*scale) @ Tk^T  (64x16, K=1568), softmax(16),
// XC = P @ Tv (64x1568).  xc written in (B_,N,C) layout (pure reshape).
// ---------------------------------------------------------------------------
__global__ __launch_bounds__(128) void k_template_attn(
    const float* __restrict__ x, const float* __restrict__ tpl,
    float* __restrict__ xc) {
  const int bh = blockIdx.x;            // 0..191
  const int b = bh / HEADS, h = bh - b * HEADS;
  const int wave = threadIdx.x >> 5;    // m-tile 0..3
  const int lane = threadIdx.x & 31;
  const int hl = lane >> 4, ln = lane & 15;

  const float* xrow =
      x + (size_t)(b * NWIN + wave * 16 + ln) * (NTOK * CDIM) + h * DDIM;
  const float* tkrow =
      tpl + (size_t)(((1 * B_IMG + b) * HEADS + h) * NWIN_T + ln) * DDIM;
  const float* tvb =
      tpl + (size_t)(((2 * B_IMG + b) * HEADS + h) * NWIN_T) * DDIM;

  // ---- logits GEMM: M=64(win), N=16(tmpl), K=1568 ----
  v8f acc0 = {}, acc1 = {};
  for (int kk = 0; kk < DDIM; kk += 8) {
    int d0 = kk + hl * 2;
    v2f a0 = *(const v2f*)(xrow + d0);
    v2f b0 = *(const v2f*)(tkrow + d0);
    acc0 = WMMA_F32(a0, b0, acc0);
    v2f a1 = *(const v2f*)(xrow + d0 + 4);
    v2f b1 = *(const v2f*)(tkrow + d0 + 4);
    acc1 = WMMA_F32(a1, b1, acc1);
  }

  // ---- softmax over 16 templates (row = one 16-lane half per VGPR) ----
  __shared__ float lp[4][16 * 18];
#pragma unroll
  for (int r = 0; r < 8; ++r) {
    float s = (acc0[r] + acc1[r]) * SCALE;
    float mx = s;
#pragma unroll
    for (int off = 8; off >= 1; off >>= 1) mx = fmaxf(mx, __shfl_xor(mx, off, 16));
    float e = __expf(s - mx);
    float sum = e;
#pragma unroll
    for (int off = 8; off >= 1; off >>= 1) sum += __shfl_xor(sum, off, 16);
    lp[wave][(r + 8 * hl) * 18 + ln] = e / sum;   // C-layout -> LDS
  }
  __syncthreads();

  // reload P in A-layout (m = ln, k = template)
  v2f af[4];
#pragma unroll
  for (int j = 0; j < 4; ++j)
    af[j] = *(const v2f*)(&lp[wave][ln * 18 + j * 4 + hl * 2]);

  // ---- XC = P @ Tv: M=64, N=1568 (98 tiles), K=16 ----
  for (int dt = 0; dt < DDIM / 16; ++dt) {
    const int dbase = dt * 16;
    v8f o = {};
#pragma unroll
    for (int j = 0; j < 4; ++j) {
      int t0 = j * 4 + hl * 2;
      v2f bb;
      bb.x = tvb[(size_t)t0 * DDIM + dbase + ln];
      bb.y = tvb[(size_t)(t0 + 1) * DDIM + dbase + ln];
      o = WMMA_F32(af[j], bb, o);
    }
#pragma unroll
    for (int r = 0; r < 8; ++r) {
      int w = wave * 16 + r + 8 * hl;
      xc[(size_t)(b * NWIN + w) * (NTOK * CDIM) + h * DDIM + dbase + ln] = o[r];
    }
  }
}

// ---------------------------------------------------------------------------
// Kernel 2/5: out[t][n] = A[t]@W[n] + bias[n] (+ res[t][n]).  N=192.
// Block = 8 waves: 128M x 64N tile; W slab staged in LDS (stride 196).
// ---------------------------------------------------------------------------
__global__ __launch_bounds__(256) void k_proj(
    const float* __restrict__ A, const float* __restrict__ W,
    const float* __restrict__ bias, const float* __restrict__ res,
    float* __restrict__ out) {
  __shared__ float lw[64 * 196];
  const int nbase = blockIdx.y * 64;
  const int mbase = blockIdx.x * 128;
  for (int idx = threadIdx.x; idx < 64 * 48; idx += 256) {
    int nl = idx / 48, kq = (idx - nl * 48) * 4;
    *(v4f*)(&lw[nl * 196 + kq]) = *(const v4f*)(W + (size_t)(nbase + nl) * CDIM + kq);
  }
  __syncthreads();

  const int wave = threadIdx.x >> 5, lane = threadIdx.x & 31;
  const int hl = lane >> 4, ln = lane & 15;
  const float* arow = A + (size_t)(mbase + wave * 16 + ln) * CDIM;

  v8f c[4] = {{}, {}, {}, {}};
  for (int kk = 0; kk < CDIM; kk += 4) {
    int k0 = kk + hl * 2;
    v2f a = *(const v2f*)(arow + k0);
#pragma unroll
    for (int j = 0; j < 4; ++j) {
      v2f bb = *(const v2f*)(&lw[(ln + 16 * j) * 196 + k0]);
      c[j] = WMMA_F32(a, bb, c[j]);
    }
  }
#pragma unroll
  for (int r = 0; r < 8; ++r) {
    int t = mbase + wave * 16 + r + 8 * hl;
    size_t ro = (size_t)t * CDIM;
#pragma unroll
    for (int j = 0; j < 4; ++j) {
      int n = nbase + j * 16 + ln;
      float val = c[j][r] + bias[n];
      if (res) val += res[ro + n];
      out[ro + n] = val;
    }
  }
}

// ---------------------------------------------------------------------------
// Kernel 3: qkv GEMM (N=576) + bias + q-scaling, scattered to (win,h,tok,hd).
// ---------------------------------------------------------------------------
__global__ __launch_bounds__(256) void k_qkv(
    const float* __restrict__ A, const float* __restrict__ W,
    const float* __restrict__ bias, float* __restrict__ qws,
    float* __restrict__ kws, float* __restrict__ vws) {
  __shared__ float lw[64 * 196];
  const int nbase = blockIdx.y * 64;
  const int mbase = blockIdx.x * 128;
  for (int idx = threadIdx.x; idx < 64 * 48; idx += 256) {
    int nl = idx / 48, kq = (idx - nl * 48) * 4;
    *(v4f*)(&lw[nl * 196 + kq]) = *(const v4f*)(W + (size_t)(nbase + nl) * CDIM + kq);
  }
  __syncthreads();

  const int wave = threadIdx.x >> 5, lane = threadIdx.x & 31;
  const int hl = lane >> 4, ln = lane & 15;
  const float* arow = A + (size_t)(mbase + wave * 16 + ln) * CDIM;

  v8f c[4] = {{}, {}, {}, {}};
  for (int kk = 0; kk < CDIM; kk += 4) {
    int k0 = kk + hl * 2;
    v2f a = *(const v2f*)(arow + k0);
#pragma unroll
    for (int j = 0; j < 4; ++j) {
      v2f bb = *(const v2f*)(&lw[(ln + 16 * j) * 196 + k0]);
      c[j] = WMMA_F32(a, bb, c[j]);
    }
  }
#pragma unroll
  for (int r = 0; r < 8; ++r) {
    int t = mbase + wave * 16 + r + 8 * hl;
    int win = t / NTOK, tok = t - win * NTOK;
#pragma unroll
    for (int j = 0; j < 4; ++j) {
      int f = nbase + j * 16 + ln;
      int which = f / CDIM;
      int fh = f - which * CDIM;
      int hh = fh >> 5, dh = fh & 31;
      float val = c[j][r] + bias[f];
      if (which == 0) val *= SCALE;
      float* dst = (which == 0) ? qws : (which == 1) ? kws : vws;
      dst[((size_t)(win * HEADS + hh) * NTOK + tok) * HD + dh] = val;
    }
  }
}

// ---------------------------------------------------------------------------
// Kernel 4: windowed self-attention, one block per (window, head).
// 49 tokens padded to 64; 4 waves = 4 m-tiles of 16 rows.
// ---------------------------------------------------------------------------
__global__ __launch_bounds__(128) void k_win_attn(
    const float* __restrict__ qws, const float* __restrict__ kws,
    const float* __restrict__ vws, const float* __restrict__ biasx,
    const float* __restrict__ mask, float* __restrict__ x2) {
  const int wh = blockIdx.x;            // win*6 + h
  const int win = wh / HEADS, h = wh - win * HEADS;
  const int wave = threadIdx.x >> 5, lane = threadIdx.x & 31;
  const int hl = lane >> 4, ln = lane & 15;

  const float* qb = qws + (size_t)wh * NTOK * HD;
  const float* kb = kws + (size_t)wh * NTOK * HD;
  const float* vb = vws + (size_t)wh * NTOK * HD;

  // Q fragments (rows indexed by ln in A-layout)
  v2f aq[8];
  const int m = wave * 16 + ln;
  const bool mval = (m < NTOK);
#pragma unroll
  for (int j = 0; j < 8; ++j) {
    v2f a = {0.f, 0.f};
    if (mval) a = *(const v2f*)(qb + m * HD + j * 4 + hl * 2);
    aq[j] = a;
  }

  // S = Q @ K^T  (64x64, K=32)
  v8f s[4];
#pragma unroll
  for (int nt = 0; nt < 4; ++nt) {
    v8f c = {};
    int n = nt * 16 + ln;
    bool nval = (n < NTOK);
#pragma unroll
    for (int j = 0; j < 8; ++j) {
      v2f bb = {0.f, 0.f};
      if (nval) bb = *(const v2f*)(kb + n * HD + j * 4 + hl * 2);
      c = WMMA_F32(aq[j], bb, c);
    }
    s[nt] = c;
  }

  // + rel-pos bias + window mask; pad columns -> -inf
  const float* bh = biasx + (size_t)h * NTOK * NTOK;
  const float* mk = mask + (size_t)(win % NWIN) * NTOK * NTOK;
#pragma unroll
  for (int nt = 0; nt < 4; ++nt) {
    int n = nt * 16 + ln;
#pragma unroll
    for (int r = 0; r < 8; ++r) {
      int mm = wave * 16 + r + 8 * hl;
      if (n < NTOK) {
        if (mm < NTOK) s[nt][r] += bh[mm * NTOK + n] + mk[mm * NTOK + n];
      } else {
        s[nt][r] = -1e30f;
      }
    }
  }

  // softmax over 64 (4 tiles x 16 lanes per row)
#pragma unroll
  for (int r = 0; r < 8; ++r) {
    float mx = fmaxf(fmaxf(s[0][r], s[1][r]), fmaxf(s[2][r], s[3][r]));
#pragma unroll
    for (int off = 8; off >= 1; off >>= 1) mx = fmaxf(mx, __shfl_xor(mx, off, 16));
    float e0 = __expf(s[0][r] - mx), e1 = __expf(s[1][r] - mx);
    float e2 = __expf(s[2][r] - mx), e3 = __expf(s[3][r] - mx);
    float sum = e0 + e1 + e2 + e3;
#pragma unroll
    for (int off = 8; off >= 1; off >>= 1) sum += __shfl_xor(sum, off, 16);
    float inv = 1.0f / sum;
    s[0][r] = e0 * inv; s[1][r] = e1 * inv;
    s[2][r] = e2 * inv; s[3][r] = e3 * inv;
  }

  // P through LDS: C-layout -> A-layout
  __shared__ float lp[4][16 * 66];
#pragma unroll
  for (int nt = 0; nt < 4; ++nt)
#pragma unroll
    for (int r = 0; r < 8; ++r)
      lp[wave][(r + 8 * hl) * 66 + nt * 16 + ln] = s[nt][r];
  __syncthreads();

  // O = P @ V  (16x32 per wave, K=64)
  v8f o0 = {}, o1 = {};
  for (int kk = 0; kk < 64; kk += 4) {
    int k0 = kk + hl * 2;
    v2f a = *(const v2f*)(&lp[wave][ln * 66 + k0]);
    v2f b0 = {0.f, 0.f}, b1 = {0.f, 0.f};
    if (k0 < NTOK)     { b0.x = vb[k0 * HD + ln];       b1.x = vb[k0 * HD + 16 + ln]; }
    if (k0 + 1 < NTOK) { b0.y = vb[(k0 + 1) * HD + ln]; b1.y = vb[(k0 + 1) * HD + 16 + ln]; }
    o0 = WMMA_F32(a, b0, o0);
    o1 = WMMA_F32(a, b1, o1);
  }
#pragma unroll
  for (int r = 0; r < 8; ++r) {
    int mm = wave * 16 + r + 8 * hl;
    if (mm < NTOK) {
      size_t off = (size_t)(win * NTOK + mm) * CDIM + h * HD;
      x2[off + ln] = o0[r];
      x2[off + 16 + ln] = o1[r];
    }
  }
}

// ---------------------------------------------------------------------------
extern "C" void kernel_launch(void* const* d_in, const int* in_sizes, int n_in,
                              void* d_out, int out_size, void* d_ws,
                              size_t ws_size, hipStream_t stream) {
  const float* x      = (const float*)d_in[0];
  const float* tpl    = (const float*)d_in[1];
  const float* mask   = (const float*)d_in[2];
  const float* table  = (const float*)d_in[3];
  const float* qkv_w  = (const float*)d_in[4];
  const float* qkv_b  = (const float*)d_in[5];
  const float* proj_w = (const float*)d_in[6];
  const float* proj_b = (const float*)d_in[7];
  const int*   ridx   = (const int*)d_in[8];
  float* out = (float*)d_out;

  const size_t SZ = (size_t)TOK * CDIM;   // 19,267,584 floats
  float* ws = (float*)d_ws;
  float* r0 = ws;            // xc, later reused for q
  float* r1 = ws + SZ;       // x1, later reused for attention output
  float* r2 = ws + 2 * SZ;   // k
  float* r3 = ws + 3 * SZ;   // v
  float* biasx = ws + 4 * SZ;

  k_bias_expand<<<dim3((HEADS * NTOK * NTOK + 255) / 256), 256, 0, stream>>>(
      table, ridx, biasx);
  k_template_attn<<<dim3(B_IMG * HEADS), 128, 0, stream>>>(x, tpl, r0);
  // x1 = x + xc @ proj_w^T + proj_b
  k_proj<<<dim3(TOK / 128, CDIM / 64), 256, 0, stream>>>(r0, proj_w, proj_b, x, r1);
  // qkv (q scaled), scattered per head; q overwrites xc region
  k_qkv<<<dim3(TOK / 128, 576 / 64), 256, 0, stream>>>(r1, qkv_w, qkv_b, r0, r2, r3);
  // attention output overwrites x1 region
  k_win_attn<<<dim3(BWIN * HEADS), 128, 0, stream>>>(r0, r2, r3, biasx, mask, r1);
  // final projection
  k_proj<<<dim3(TOK / 128, CDIM / 64), 256, 0, stream>>>(r1, proj_w, proj_b, nullptr, out);
}